// TaskAlignedAssigner_83940840833737
// MI455X (gfx1250) — compile-verified
//
#include <hip/hip_runtime.h>
#include <hip/hip_bf16.h>

#define TOPK 10
#define NUM_CLASSES 80
#define EPSF 1e-9f

typedef __attribute__((ext_vector_type(2))) float v2f;
typedef __attribute__((ext_vector_type(8))) float v8f;

// ---------------------------------------------------------------------------
// K1: per (b, n, a): IoU, score gather, align metric, in-gt mask
// ---------------------------------------------------------------------------
__global__ void metric_kernel(const float* __restrict__ pd_scores,
                              const float* __restrict__ pd_bboxes,
                              const float* __restrict__ anc,
                              const int* __restrict__ gt_labels,
                              const float* __restrict__ gt_bboxes,
                              float* __restrict__ overlaps,
                              float* __restrict__ alignm,
                              float* __restrict__ metrics,
                              int BS_, int N_, int A_, int C_) {
    size_t i = (size_t)blockIdx.x * blockDim.x + threadIdx.x;
    size_t tot = (size_t)BS_ * N_ * A_;
    if (i >= tot) return;
    int a = (int)(i % A_);
    int r = (int)(i / A_);      // r = b*N + j
    int b = r / N_;

    const float* gb = gt_bboxes + (size_t)r * 4;
    float g0 = gb[0], g1 = gb[1], g2 = gb[2], g3 = gb[3];
    const float* pb = pd_bboxes + ((size_t)b * A_ + a) * 4;
    float p0 = pb[0], p1 = pb[1], p2 = pb[2], p3 = pb[3];

    float ow = fmaxf(fminf(g2, p2) - fmaxf(g0, p0), 0.0f);
    float oh = fmaxf(fminf(g3, p3) - fmaxf(g1, p1), 0.0f);
    float ov = ow * oh;
    float ar1 = fmaxf(g2 - g0, 0.0f) * fmaxf(g3 - g1, 0.0f);
    float ar2 = fmaxf(p2 - p0, 0.0f) * fmaxf(p3 - p1, 0.0f);
    float iou = ov / (ar1 + ar2 - ov + EPSF);

    int lbl = gt_labels[r];
    float sc = pd_scores[((size_t)b * A_ + a) * C_ + lbl];
    float i2 = iou * iou;
    float i6 = i2 * i2 * i2;               // iou ** 6
    float al = sqrtf(sc) * i6;             // score ** 0.5 * iou ** 6

    float ax = anc[(size_t)a * 2 + 0];
    float ay = anc[(size_t)a * 2 + 1];
    float dmin = fminf(fminf(ax - g0, ay - g1), fminf(g2 - ax, g3 - ay));
    float mask = (dmin > EPSF) ? 1.0f : 0.0f;

    overlaps[i] = iou;
    alignm[i]   = al;
    metrics[i]  = al * mask;
}

// ---------------------------------------------------------------------------
// K2: per (b, n) row: top-10 over A with lowest-index tie-break, then write
// mask_pos = is_in_topk * mask_in_gts  (metrics buffer is overwritten in place)
// ---------------------------------------------------------------------------
__global__ void topk_kernel(const float* __restrict__ metrics_in,
                            const float* __restrict__ gt_bboxes,
                            const float* __restrict__ anc,
                            float* __restrict__ mpos,
                            int N_, int A_) {
    extern __shared__ float sm[];          // A_ floats: cached metrics row
    __shared__ float rv[256];
    __shared__ int   ri[256];
    __shared__ int   sel[TOPK];
    __shared__ float selv[TOPK];
    __shared__ float gbox[4];

    int row = blockIdx.x;                  // row = b*N + j
    const float* mrow = metrics_in + (size_t)row * A_;
    for (int a = threadIdx.x; a < A_; a += blockDim.x) sm[a] = mrow[a];
    if (threadIdx.x < 4) gbox[threadIdx.x] = gt_bboxes[(size_t)row * 4 + threadIdx.x];
    __syncthreads();

    for (int t = 0; t < TOPK; ++t) {
        float bv = -1.0f; int bi = 0x7fffffff;
        for (int a = threadIdx.x; a < A_; a += blockDim.x) {
            float v = sm[a];
            if (v > bv) { bv = v; bi = a; }   // strict > keeps lowest index
        }
        rv[threadIdx.x] = bv; ri[threadIdx.x] = bi;
        __syncthreads();
        for (int s = blockDim.x >> 1; s > 0; s >>= 1) {
            if (threadIdx.x < (unsigned)s) {
                float v2 = rv[threadIdx.x + s]; int idx2 = ri[threadIdx.x + s];
                if (v2 > rv[threadIdx.x] ||
                    (v2 == rv[threadIdx.x] && idx2 < ri[threadIdx.x])) {
                    rv[threadIdx.x] = v2; ri[threadIdx.x] = idx2;
                }
            }
            __syncthreads();
        }
        if (threadIdx.x == 0) {
            sel[t] = ri[0]; selv[t] = rv[0];
            sm[ri[0]] = -2.0f;             // remove from further selection
        }
        __syncthreads();
    }

    float* orow = mpos + (size_t)row * A_;
    for (int a = threadIdx.x; a < A_; a += blockDim.x) orow[a] = 0.0f;
    __syncthreads();

    bool valid = selv[0] > EPSF;           // topk_vals.max() > eps
    if (valid && threadIdx.x < TOPK) {
        int a = sel[threadIdx.x];
        float ax = anc[(size_t)a * 2 + 0];
        float ay = anc[(size_t)a * 2 + 1];
        float dmin = fminf(fminf(ax - gbox[0], ay - gbox[1]),
                           fminf(gbox[2] - ax, gbox[3] - ay));
        if (dmin > EPSF) orow[a] = 1.0f;   // * mask_in_gts
    }
}

// ---------------------------------------------------------------------------
// K3a: fg_pre[b,a] = sum_n mask_pos[b,n,a] via chained V_WMMA_F32_16X16X4_F32
// A-matrix = ones(16x4)  ->  every row of D holds the column sums of B, so
// the result is independent of the K-slot permutation within the B fragment.
// One wave handles one (b, 16-anchor tile); uniform wave control flow keeps
// EXEC all-ones around the WMMA as the ISA requires.
// ---------------------------------------------------------------------------
__global__ void fg_pre_wmma(const float* __restrict__ mpos,
                            float* __restrict__ fg_pre,
                            int BS_, int N_, int A_) {
    int wave = blockIdx.x * (blockDim.x >> 5) + (threadIdx.x >> 5);
    int lane = threadIdx.x & 31;
    int tiles = (A_ + 15) >> 4;
    if (wave >= BS_ * tiles) return;       // wave-uniform
    int b = wave / tiles;
    int tile = wave % tiles;
    int a0 = tile * 16;
    const float* base = mpos + (size_t)b * N_ * A_;
    int col = lane & 15;
    int half = lane >> 4;
    int a = a0 + col;

    if (a0 + 16 <= A_) {
        v2f av; av.x = 1.0f; av.y = 1.0f;  // ones A-matrix
        v8f c = {};
        int kmax = (N_ >> 2) << 2;
        for (int kk = 0; kk < kmax; kk += 4) {
            v2f bv;
            bv.x = base[(size_t)(kk + half * 2 + 0) * A_ + a];
            bv.y = base[(size_t)(kk + half * 2 + 1) * A_ + a];
            c = __builtin_amdgcn_wmma_f32_16x16x4_f32(
                    false, av, false, bv, (short)0, c, false, false);
        }
        float s = c[0];                    // every lane: colsum[lane & 15]
        for (int nn = kmax; nn < N_; ++nn) // tail rows (none when N_%4==0)
            s += base[(size_t)nn * A_ + a];
        if (lane < 16) fg_pre[(size_t)b * A_ + a] = s;
    } else {
        if (lane < 16 && a < A_) {
            float s = 0.0f;
            for (int nn = 0; nn < N_; ++nn) s += base[(size_t)nn * A_ + a];
            fg_pre[(size_t)b * A_ + a] = s;
        }
    }
}

// ---------------------------------------------------------------------------
// K3b: per (b, a): resolve multi-assignment with overlaps argmax; produce
// final mask_pos column, fg_mask and target_gt_idx (= first n with mask==1)
// ---------------------------------------------------------------------------
__global__ void resolve_kernel(const float* __restrict__ overlaps,
                               float* __restrict__ mpos,
                               const float* __restrict__ fg_pre,
                               float* __restrict__ fg_mask,
                               int* __restrict__ tgt,
                               int N_, int A_, int total) {
    int i = blockIdx.x * blockDim.x + threadIdx.x;
    if (i >= total) return;
    int b = i / A_;
    int a = i % A_;
    size_t col = (size_t)b * N_ * A_ + a;

    float fgp = fg_pre[i];
    int t; float fg;
    if (fgp > 1.0f) {                      // anchor claimed by multiple GTs
        float bv = -1.0f; int bi = 0;
        for (int nn = 0; nn < N_; ++nn) {
            float v = overlaps[col + (size_t)nn * A_];
            if (v > bv) { bv = v; bi = nn; }
        }
        for (int nn = 0; nn < N_; ++nn)
            mpos[col + (size_t)nn * A_] = (nn == bi) ? 1.0f : 0.0f;
        t = bi; fg = 1.0f;
    } else {
        t = 0; fg = fgp;
        for (int nn = 0; nn < N_; ++nn) {  // argmax of 0/1 column = first 1
            if (mpos[col + (size_t)nn * A_] > 0.5f) { t = nn; break; }
        }
    }
    tgt[i] = t;
    fg_mask[i] = fg;
}

// ---------------------------------------------------------------------------
// K4: per (b, n): pos_align = max_a(align*mask), pos_over = max_a(ov*mask)
// ---------------------------------------------------------------------------
__global__ void rowmax_kernel(const float* __restrict__ alignm,
                              const float* __restrict__ overlaps,
                              const float* __restrict__ mpos,
                              float* __restrict__ pos_align,
                              float* __restrict__ pos_over,
                              int A_) {
    __shared__ float sa[256];
    __shared__ float so[256];
    int row = blockIdx.x;
    const float* ar = alignm   + (size_t)row * A_;
    const float* orr = overlaps + (size_t)row * A_;
    const float* mr = mpos     + (size_t)row * A_;
    float ma = 0.0f, mo = 0.0f;            // all terms are >= 0
    for (int a = threadIdx.x; a < A_; a += blockDim.x) {
        float m = mr[a];
        ma = fmaxf(ma, ar[a] * m);
        mo = fmaxf(mo, orr[a] * m);
    }
    sa[threadIdx.x] = ma; so[threadIdx.x] = mo;
    __syncthreads();
    for (int s = blockDim.x >> 1; s > 0; s >>= 1) {
        if (threadIdx.x < (unsigned)s) {
            sa[threadIdx.x] = fmaxf(sa[threadIdx.x], sa[threadIdx.x + s]);
            so[threadIdx.x] = fmaxf(so[threadIdx.x], so[threadIdx.x + s]);
        }
        __syncthreads();
    }
    if (threadIdx.x == 0) { pos_align[row] = sa[0]; pos_over[row] = so[0]; }
}

// ---------------------------------------------------------------------------
// K5: per (b, a): norm, gather label/bbox, write all four outputs (as float)
// ---------------------------------------------------------------------------
__global__ void finalize_kernel(const float* __restrict__ alignm,
                                const float* __restrict__ mpos,
                                const float* __restrict__ pos_align,
                                const float* __restrict__ pos_over,
                                const int* __restrict__ gt_labels,
                                const float* __restrict__ gt_bboxes,
                                const int* __restrict__ tgt,
                                const float* __restrict__ fg_mask,
                                float* __restrict__ out_labels,
                                float* __restrict__ out_bboxes,
                                float* __restrict__ out_scores,
                                float* __restrict__ out_fg,
                                int N_, int A_, int C_, int total) {
    int i = blockIdx.x * blockDim.x + threadIdx.x;
    if (i >= total) return;
    int b = i / A_;
    int a = i % A_;
    size_t col = (size_t)b * N_ * A_ + a;

    float norm = 0.0f;
    for (int nn = 0; nn < N_; ++nn) {
        float am = alignm[col + (size_t)nn * A_] * mpos[col + (size_t)nn * A_];
        float term = am * pos_over[b * N_ + nn] / (pos_align[b * N_ + nn] + EPSF);
        norm = fmaxf(norm, term);
    }

    int t = tgt[i];
    int lbl = gt_labels[b * N_ + t];
    if (lbl < 0) lbl = 0;
    bool fg = fg_mask[i] > 0.0f;

    out_labels[i] = (float)lbl;
    const float* gbb = gt_bboxes + ((size_t)b * N_ + t) * 4;
    float* ob = out_bboxes + (size_t)i * 4;
    ob[0] = gbb[0]; ob[1] = gbb[1]; ob[2] = gbb[2]; ob[3] = gbb[3];
    float* os = out_scores + (size_t)i * C_;
    for (int c = 0; c < C_; ++c)
        os[c] = (fg && c == lbl) ? norm : 0.0f;
    out_fg[i] = fg ? 1.0f : 0.0f;
}

// ---------------------------------------------------------------------------
extern "C" void kernel_launch(void* const* d_in, const int* in_sizes, int n_in,
                              void* d_out, int out_size, void* d_ws, size_t ws_size,
                              hipStream_t stream) {
    const float* pd_scores = (const float*)d_in[0];   // (bs, A, C)
    const float* pd_bboxes = (const float*)d_in[1];   // (bs, A, 4)
    const float* anc       = (const float*)d_in[2];   // (A, 2)
    const int*   gt_labels = (const int*)d_in[3];     // (bs, n, 1)
    const float* gt_bboxes = (const float*)d_in[4];   // (bs, n, 4)

    const int C_  = NUM_CLASSES;
    const int A_  = in_sizes[2] / 2;
    const int BS_ = in_sizes[0] / (A_ * C_);
    const int N_  = in_sizes[3] / BS_;

    // workspace layout (floats): ~3 * bs*n*A + small
    size_t nA = (size_t)BS_ * N_ * A_;
    size_t BA = (size_t)BS_ * A_;
    float* ws        = (float*)d_ws;
    float* overlaps  = ws;
    float* alignm    = ws + nA;
    float* mpos      = ws + 2 * nA;        // metrics, then overwritten by mask_pos
    float* fg_pre    = ws + 3 * nA;
    float* fg_mask   = fg_pre + BA;
    int*   tgt       = (int*)(fg_mask + BA);
    float* pos_align = (float*)(tgt + BA);
    float* pos_over  = pos_align + (size_t)BS_ * N_;

    // output layout: labels | bboxes | scores | fg  (all as float)
    float* out_labels = (float*)d_out;
    float* out_bboxes = out_labels + BA;
    float* out_scores = out_bboxes + BA * 4;
    float* out_fg     = out_scores + BA * C_;

    // K1
    {
        int blocks = (int)((nA + 255) / 256);
        metric_kernel<<<blocks, 256, 0, stream>>>(
            pd_scores, pd_bboxes, anc, gt_labels, gt_bboxes,
            overlaps, alignm, mpos, BS_, N_, A_, C_);
    }
    // K2: one block per (b, n) row, row cached in dynamic LDS
    topk_kernel<<<BS_ * N_, 256, (size_t)A_ * sizeof(float), stream>>>(
        mpos, gt_bboxes, anc, mpos, N_, A_);
    // K3a: WMMA column sums (one wave per 16-anchor tile)
    {
        int tiles = (A_ + 15) >> 4;
        int waves = BS_ * tiles;
        int blocks = (waves + 7) / 8;      // 8 waves per 256-thread block
        fg_pre_wmma<<<blocks, 256, 0, stream>>>(mpos, fg_pre, BS_, N_, A_);
    }
    // K3b
    {
        int total = (int)BA;
        resolve_kernel<<<(total + 255) / 256, 256, 0, stream>>>(
            overlaps, mpos, fg_pre, fg_mask, tgt, N_, A_, total);
    }
    // K4
    rowmax_kernel<<<BS_ * N_, 256, 0, stream>>>(
        alignm, overlaps, mpos, pos_align, pos_over, A_);
    // K5
    {
        int total = (int)BA;
        finalize_kernel<<<(total + 255) / 256, 256, 0, stream>>>(
            alignm, mpos, pos_align, pos_over, gt_labels, gt_bboxes,
            tgt, fg_mask, out_labels, out_bboxes, out_scores, out_fg,
            N_, A_, C_, total);
    }
}